// graph_node_update_2302102471102
// MI455X (gfx1250) — compile-verified
//
#include <hip/hip_runtime.h>
#include <hip/hip_bf16.h>

typedef __attribute__((ext_vector_type(2))) float v2f;
typedef __attribute__((ext_vector_type(8))) float v8f;

#define D 128          // D_IN == D_OUT == 128
#define LN_EPS 1e-5f

// ---------------------------------------------------------------------------
// Degree kernels: deg[i] = 1 (self loop) + #incoming edges; then dinv=rsqrt.
// ---------------------------------------------------------------------------
__global__ void deg_init_kernel(float* __restrict__ deg, int N) {
    int i = blockIdx.x * blockDim.x + threadIdx.x;
    if (i < N) deg[i] = 1.0f;
}

__global__ void deg_count_kernel(const int* __restrict__ dst,
                                 float* __restrict__ deg, int E) {
    int e = blockIdx.x * blockDim.x + threadIdx.x;
    if (e < E) atomicAdd(&deg[dst[e]], 1.0f);
}

__global__ void deg_rsqrt_kernel(float* __restrict__ deg, int N) {
    int i = blockIdx.x * blockDim.x + threadIdx.x;
    if (i < N) deg[i] = rsqrtf(deg[i]);   // deg >= 1 always (self loop)
}

// ---------------------------------------------------------------------------
// Dual GEMM via fp32 WMMA (V_WMMA_F32_16X16X4_F32), wave32.
//   block = 256 threads = 8 waves; one 16-row block of x per block.
//   waves 0..3 : h   = x @ W_gcn            -> ws h buffer
//   waves 4..7 : acc = x @ W_lin + b + 1e-6 -> d_out (accumulator)
//   each wave owns a 16x32 slice (two 16x16 WMMA accumulators), K-loop 128/4.
// A 16x4 f32 layout: lanes 0-15 = rows, VGPR0/1 = K{0,1} (lanes 0-15),
// K{2,3} (lanes 16-31). B 4x16: rows striped across lanes within VGPR.
// ---------------------------------------------------------------------------
__global__ void gemm_dual_kernel(const float* __restrict__ x,
                                 const float* __restrict__ Wg,
                                 const float* __restrict__ Wl,
                                 const float* __restrict__ bg,
                                 float* __restrict__ h,
                                 float* __restrict__ acc, int N) {
    const int wave = threadIdx.x >> 5;
    const int lane = threadIdx.x & 31;
    const int l16  = lane & 15;
    const int lhalf = lane >> 4;              // 0: K=0,1   1: K=2,3
    const int rowBase = blockIdx.x * 16;
    const int mat = wave >> 2;                // 0 -> W_gcn, 1 -> W_lin
    const int colBase = (wave & 3) * 32;
    const float* __restrict__ W = mat ? Wl : Wg;

    int arow = rowBase + l16;
    if (arow >= N) arow = N - 1;              // clamp OOB reads (N%16==0 anyway)
    const float* __restrict__ xrow = x + (size_t)arow * D;

    v8f c0 = {};
    v8f c1 = {};
    for (int k0 = 0; k0 < D; k0 += 4) {
        const int ka = k0 + lhalf * 2;
        v2f a;
        a.x = xrow[ka];
        a.y = xrow[ka + 1];
        v2f b0, b1;
        b0.x = W[(size_t)ka       * D + colBase + l16];
        b0.y = W[(size_t)(ka + 1) * D + colBase + l16];
        b1.x = W[(size_t)ka       * D + colBase + 16 + l16];
        b1.y = W[(size_t)(ka + 1) * D + colBase + 16 + l16];
        c0 = __builtin_amdgcn_wmma_f32_16x16x4_f32(false, a, false, b0,
                                                   (short)0, c0, false, false);
        c1 = __builtin_amdgcn_wmma_f32_16x16x4_f32(false, a, false, b1,
                                                   (short)0, c1, false, false);
    }

    // C/D layout: VGPR r -> M = r + 8*lhalf, N = l16 (per 16-col tile)
    #pragma unroll
    for (int r = 0; r < 8; ++r) {
        const int row = rowBase + r + lhalf * 8;
        if (row < N) {
            const int col0 = colBase + l16;
            const int col1 = colBase + 16 + l16;
            if (mat == 0) {
                h[(size_t)row * D + col0] = c0[r];
                h[(size_t)row * D + col1] = c1[r];
            } else {
                acc[(size_t)row * D + col0] = c0[r] + bg[col0] + 1e-6f;
                acc[(size_t)row * D + col1] = c1[r] + bg[col1] + 1e-6f;
            }
        }
    }
}

// ---------------------------------------------------------------------------
// Edge scatter: task t < E is a real edge, task t >= E is self-loop (t-E).
// 128 threads per task (one feature each): coalesced 512B gather of h[src],
// scale by dinv[src]*dinv[dst], f32 atomic add into acc[dst].
// ---------------------------------------------------------------------------
__global__ void scatter_kernel(const int* __restrict__ src,
                               const int* __restrict__ dst,
                               const float* __restrict__ dinv,
                               const float* __restrict__ h,
                               float* __restrict__ acc,
                               int E, int N) {
    const long long task = (long long)blockIdx.x * 2 + (threadIdx.x >> 7);
    const int f = threadIdx.x & (D - 1);
    const long long total = (long long)E + N;
    if (task >= total) return;
    int s, d;
    if (task < E) { s = src[task]; d = dst[task]; }
    else          { s = d = (int)(task - E); }
    const float norm = dinv[s] * dinv[d];
    const float v = h[(size_t)s * D + f] * norm;
    atomicAdd(&acc[(size_t)d * D + f], v);
}

// ---------------------------------------------------------------------------
// LayerNorm: one wave32 per row, float4 per lane, shfl_xor reduction.
// In-place safe (each lane reads then rewrites only its own float4).
// ---------------------------------------------------------------------------
__global__ void ln_kernel(const float* __restrict__ acc,
                          const float* __restrict__ lnw,
                          const float* __restrict__ lnb,
                          float* __restrict__ out, int N) {
    const int row  = blockIdx.x * 8 + (threadIdx.x >> 5);
    const int lane = threadIdx.x & 31;
    if (row >= N) return;

    const float4 v = ((const float4*)(acc + (size_t)row * D))[lane];
    float s  = v.x + v.y + v.z + v.w;
    float sq = v.x * v.x + v.y * v.y + v.z * v.z + v.w * v.w;
    #pragma unroll
    for (int off = 16; off > 0; off >>= 1) {
        s  += __shfl_xor(s,  off, 32);
        sq += __shfl_xor(sq, off, 32);
    }
    const float mu  = s * (1.0f / (float)D);
    const float var = sq * (1.0f / (float)D) - mu * mu;
    const float inv = rsqrtf(var + LN_EPS);

    const float4 w = ((const float4*)lnw)[lane];
    const float4 b = ((const float4*)lnb)[lane];
    float4 o;
    o.x = (v.x - mu) * inv * w.x + b.x;
    o.y = (v.y - mu) * inv * w.y + b.y;
    o.z = (v.z - mu) * inv * w.z + b.z;
    o.w = (v.w - mu) * inv * w.w + b.w;
    ((float4*)(out + (size_t)row * D))[lane] = o;
}

// ---------------------------------------------------------------------------
extern "C" void kernel_launch(void* const* d_in, const int* in_sizes, int n_in,
                              void* d_out, int out_size, void* d_ws, size_t ws_size,
                              hipStream_t stream) {
    const int*   adj = (const int*)  d_in[0];   // (2, E)
    const float* x   = (const float*)d_in[1];   // (N, 128)
    const float* Wg  = (const float*)d_in[2];   // (128, 128)
    const float* bg  = (const float*)d_in[3];   // (128,)
    const float* Wl  = (const float*)d_in[4];   // (128, 128)
    const float* lnw = (const float*)d_in[5];   // (128,)
    const float* lnb = (const float*)d_in[6];   // (128,)

    const int E = in_sizes[0] / 2;
    const int N = in_sizes[1] / D;
    const int* src = adj;
    const int* dst = adj + E;

    // workspace: dinv (N floats, 256B aligned) | h (N*128 floats)
    float* dinv = (float*)d_ws;
    size_t off  = ((size_t)N * sizeof(float) + 255) & ~(size_t)255;
    float* h    = (float*)((char*)d_ws + off);
    float* acc  = (float*)d_out;                // accumulator lives in d_out

    // 1) degrees -> dinv
    deg_init_kernel <<<(N + 255) / 256, 256, 0, stream>>>(dinv, N);
    deg_count_kernel<<<(E + 255) / 256, 256, 0, stream>>>(dst, dinv, E);
    deg_rsqrt_kernel<<<(N + 255) / 256, 256, 0, stream>>>(dinv, N);

    // 2) dual GEMM (WMMA f32): h = x@Wg ; acc = x@Wl + bg + 1e-6
    gemm_dual_kernel<<<(N + 15) / 16, 256, 0, stream>>>(x, Wg, Wl, bg, h, acc, N);

    // 3) edge + self-loop scatter into acc
    const long long tasks = (long long)E + N;
    scatter_kernel<<<(unsigned)((tasks + 1) / 2), 256, 0, stream>>>(src, dst, dinv,
                                                                   h, acc, E, N);

    // 4) layernorm acc -> out (in place)
    ln_kernel<<<(N + 7) / 8, 256, 0, stream>>>(acc, lnw, lnb, (float*)d_out, N);
}